// TemporalGNNEncoder_850403524856
// MI455X (gfx1250) — compile-verified
//
#include <hip/hip_runtime.h>

#define H 4
#define HID 128
#define LDS_STRIDE 132   // 128 + 4 dwords TDM pad -> conflict-free A reads

typedef __attribute__((ext_vector_type(2))) float v2f;
typedef __attribute__((ext_vector_type(8))) float v8f;
typedef unsigned int u32x4 __attribute__((ext_vector_type(4)));
typedef int i32x4 __attribute__((ext_vector_type(4)));
typedef int i32x8 __attribute__((ext_vector_type(8)));

#if defined(__has_builtin)
#if __has_builtin(__builtin_amdgcn_tensor_load_to_lds)
#define USE_TDM 1
#endif
#endif
#if __has_include(<hip/amd_detail/amd_gfx1250_TDM.h>)
#define TDM_SIX_ARGS 1
#endif

// ---------------------------------------------------------------- utilities
__global__ void fill_kernel(float* __restrict__ p, float v, long n) {
    long i = (long)blockIdx.x * blockDim.x + threadIdx.x;
    long stride = (long)gridDim.x * blockDim.x;
    for (; i < n; i += stride) p[i] = v;
}

// ------------------------------------------------ input projection D -> 128
__global__ void proj_kernel(const float* __restrict__ x, const float* __restrict__ W,
                            const float* __restrict__ b, float* __restrict__ h,
                            int N, int D) {
    long t = (long)blockIdx.x * blockDim.x + threadIdx.x;
    if (t >= (long)N * HID) return;
    int n = (int)(t >> 7), o = (int)(t & 127);
    float acc = b[o];
    for (int d = 0; d < D; ++d) acc += x[(long)n * D + d] * W[d * HID + o];
    h[t] = acc;
}

#ifdef USE_TDM
// TDM: DMA a [rows<=16, 128] f32 tile (row stride 128) into LDS offset 0,
// padding LDS by 4 dwords every 128 dwords (row stride becomes 132 dwords).
__device__ __forceinline__ void tdm_load_tile(const float* gaddr, int rowsAvail) {
    unsigned long long ga = (unsigned long long)gaddr;
    u32x4 g0;
    g0[0] = 1u;                                    // count=1, user descriptor
    g0[1] = 0u;                                    // lds_addr = 0 (tile at LDS base)
    g0[2] = (unsigned)(ga & 0xFFFFFFFFu);          // global_addr[31:0]
    g0[3] = (unsigned)((ga >> 32) & 0x01FFFFFFu)   // global_addr[56:32]
          | (2u << 30);                            // type = 2 ("image")
    i32x8 g1;
    unsigned dim1 = (unsigned)rowsAvail;           // OOB rows read as zero
    g1[0] = (int)((2u << 16)        // data_size = 4 bytes
                | (1u << 20)        // pad_enable
                | (6u << 22)        // pad_interval: 128 dwords
                | (3u << 25));      // pad_amount: 4 dwords
    g1[1] = (int)(128u << 16);      // tensor_dim0 = 128 (low 16 bits at bit 48)
    g1[2] = (int)((dim1 & 0xFFFFu) << 16);                    // tensor_dim1 lo
    g1[3] = (int)(((dim1 >> 16) & 0xFFFFu) | (128u << 16));   // tensor_dim1 hi | tile_dim0=128
    g1[4] = 16;                     // tile_dim1 = 16 rows (tile_dim2 = 0)
    g1[5] = 128;                    // tensor_dim0_stride = 128 elements
    g1[6] = 0;
    g1[7] = 0;
    i32x4 z4 = {0, 0, 0, 0};
#ifdef TDM_SIX_ARGS
    i32x8 z8 = {0, 0, 0, 0, 0, 0, 0, 0};
    __builtin_amdgcn_tensor_load_to_lds(g0, g1, z4, z4, z8, 0);
#else
    __builtin_amdgcn_tensor_load_to_lds(g0, g1, z4, z4, 0);
#endif
}
#endif

// -------------------------------------- H[N,128] = X[N,128] @ W[128,128] (f32 WMMA)
// block = 256 threads = 8 waves; one 16x128 X tile staged in LDS (TDM), each
// wave owns one 16-column tile of W and emits one 16x16 D tile.
__global__ __launch_bounds__(256) void gemm128_wmma(const float* __restrict__ X,
                                                    const float* __restrict__ W,
                                                    float* __restrict__ Hout, int N) {
    __shared__ float tileX[16 * LDS_STRIDE];
    const int wave = threadIdx.x >> 5;
    const int lane = threadIdx.x & 31;
    const int m0 = blockIdx.x << 4;

#ifdef USE_TDM
    if (threadIdx.x < 32) {
        tdm_load_tile(X + (long)m0 * HID, N - m0);
        __builtin_amdgcn_s_wait_tensorcnt(0);
        asm volatile("" ::: "memory");
        if (lane == 0) {
            // The TDM intrinsic's LDS write is invisible to alias analysis;
            // without a visible store, all tileX loads fold to undef. Do an
            // opaque-index volatile read-modify-write (numerically a no-op)
            // so the compiler must emit real ds_loads for the A fragments.
            unsigned idx0 = 0;
            asm volatile("" : "+v"(idx0));
            volatile float* tp = tileX;
            tp[idx0] = tp[idx0];
        }
    }
    __syncthreads();
#else
    for (int i = threadIdx.x; i < 16 * HID; i += 256) {
        int r = i >> 7, cc = i & 127;
        int gr = m0 + r;
        tileX[r * LDS_STRIDE + cc] = (gr < N) ? X[(long)gr * HID + cc] : 0.f;
    }
    __syncthreads();
#endif

    const int n0 = wave << 4;
    const int l = lane & 15;
    const int koff = (lane >> 4) << 1;            // K offset 0 / 2 per lane half
    const float* xr = &tileX[l * LDS_STRIDE];
    v8f c = {};
#pragma unroll 4
    for (int k0 = 0; k0 < HID; k0 += 4) {
        v2f a, bm;
        a.x = xr[k0 + koff];
        a.y = xr[k0 + koff + 1];
        bm.x = W[(k0 + koff) * HID + n0 + l];
        bm.y = W[(k0 + koff + 1) * HID + n0 + l];
        c = __builtin_amdgcn_wmma_f32_16x16x4_f32(false, a, false, bm,
                                                  (short)0, c, false, false);
    }
    const int mbase = m0 + ((lane >> 4) << 3);    // lanes>=16 hold rows m0+8..m0+15
#pragma unroll
    for (int r = 0; r < 8; ++r) {
        int rr = mbase + r;
        if (rr < N) Hout[(long)rr * HID + n0 + l] = c[r];
    }
}

// ----------------------------------------- a[n,h] = sum_c h[n,h*32+c]*att[h,c]
__global__ void attdot_kernel(const float* __restrict__ Hm, const float* __restrict__ att,
                              float* __restrict__ a, int N) {
    long t = (long)blockIdx.x * blockDim.x + threadIdx.x;
    if (t >= (long)N * H) return;
    int n = (int)(t >> 2), h = (int)(t & 3);
    const float* hp = Hm + (long)n * HID + h * 32;
    const float* ap = att + h * 32;
    float s = 0.f;
#pragma unroll
    for (int c = 0; c < 32; ++c) s += hp[c] * ap[c];
    a[t] = s;
}

// ------------------------------------------------- float atomic max (int trick)
__device__ inline void atomicMaxF(float* addr, float v) {
    if (v >= 0.0f) atomicMax((int*)addr, __float_as_int(v));
    else           atomicMin((unsigned int*)addr, __float_as_uint(v));
}

// pass1: segment max of leaky_relu(a_s[src]+a_d[dst])
__global__ void edge_max_kernel(const int* __restrict__ src, const int* __restrict__ dst,
                                const float* __restrict__ as, const float* __restrict__ ad,
                                float* __restrict__ mb, int E) {
    long t = (long)blockIdx.x * blockDim.x + threadIdx.x;
    if (t >= (long)E * H) return;
    int e = (int)(t >> 2), h = (int)(t & 3);
    int s = src[e], d = dst[e];
    float v = as[s * H + h] + ad[d * H + h];
    v = v > 0.f ? v : 0.2f * v;
    atomicMaxF(&mb[d * H + h], v);
}

// pass2: e = exp(logit - m[dst]); denom[dst] += e; store e
__global__ void edge_exp_kernel(const int* __restrict__ src, const int* __restrict__ dst,
                                const float* __restrict__ as, const float* __restrict__ ad,
                                const float* __restrict__ mb, float* __restrict__ ee,
                                float* __restrict__ den, int E) {
    long t = (long)blockIdx.x * blockDim.x + threadIdx.x;
    if (t >= (long)E * H) return;
    int e = (int)(t >> 2), h = (int)(t & 3);
    int s = src[e], d = dst[e];
    float v = as[s * H + h] + ad[d * H + h];
    v = v > 0.f ? v : 0.2f * v;
    float ev = __expf(v - mb[d * H + h]);
    ee[t] = ev;
    atomicAdd(&den[d * H + h], ev);
}

// pass3: acc[dst, c] += alpha * hs[src, c]
__global__ void edge_scatter_kernel(const int* __restrict__ src, const int* __restrict__ dst,
                                    const float* __restrict__ hs, const float* __restrict__ ee,
                                    const float* __restrict__ den, float* __restrict__ acc, int E) {
    long t = (long)blockIdx.x * blockDim.x + threadIdx.x;
    if (t >= (long)E * HID) return;
    int e = (int)(t >> 7), c = (int)(t & 127);
    int h = c >> 5;
    int s = src[e], d = dst[e];
    float alpha = ee[(long)e * H + h] / (den[d * H + h] + 1e-16f);
    atomicAdd(&acc[(long)d * HID + c], alpha * hs[(long)s * HID + c]);
}

// new = (acc + sum_bias) * invk ; y = relu(new) + x ; x = layernorm(y)*g + b
__global__ void post_kernel(float* __restrict__ x, const float* __restrict__ acc,
                            const float* __restrict__ b0, const float* __restrict__ b1,
                            const float* __restrict__ b2, float invk,
                            const float* __restrict__ g, const float* __restrict__ bn, int N) {
    int n = blockIdx.x;
    int c = threadIdx.x;
    __shared__ float red[HID];
    float v = acc[(long)n * HID + c];
    float bb = 0.f;
    if (b0) bb += b0[c];
    if (b1) bb += b1[c];
    if (b2) bb += b2[c];
    v = (v + bb) * invk;
    float y = fmaxf(v, 0.f) + x[(long)n * HID + c];
    red[c] = y; __syncthreads();
    for (int s = 64; s > 0; s >>= 1) { if (c < s) red[c] += red[c + s]; __syncthreads(); }
    float mu = red[0] * (1.f / HID);
    __syncthreads();
    float d0 = y - mu;
    red[c] = d0 * d0; __syncthreads();
    for (int s = 64; s > 0; s >>= 1) { if (c < s) red[c] += red[c + s]; __syncthreads(); }
    float var = red[0] * (1.f / HID);
    x[(long)n * HID + c] = d0 * rsqrtf(var + 1e-5f) * g[c] + bn[c];
}

// global mean pool (atomic sum + count, then divide)
__global__ void pool_kernel(const float* __restrict__ x, const int* __restrict__ batch,
                            float* __restrict__ out, float* __restrict__ cnt, int N, int ntIdx) {
    long t = (long)blockIdx.x * blockDim.x + threadIdx.x;
    if (t >= (long)N * HID) return;
    int n = (int)(t >> 7), c = (int)(t & 127);
    int b = batch[n];
    atomicAdd(&out[(long)b * (4 * HID) + ntIdx * HID + c], x[t]);
    if (c == 0) atomicAdd(&cnt[ntIdx * 64 + b], 1.0f);
}

__global__ void pool_div_kernel(float* __restrict__ out, const float* __restrict__ cnt) {
    int t = blockIdx.x * blockDim.x + threadIdx.x;
    if (t >= 64 * 4 * HID) return;
    int b = t >> 9;
    int col = t & 511;
    int nt = col >> 7;
    out[t] /= fmaxf(cnt[nt * 64 + b], 1.0f);
}

// ---------------------------------------------------------------- host side
extern "C" void kernel_launch(void* const* d_in, const int* in_sizes, int n_in,
                              void* d_out, int out_size, void* d_ws, size_t ws_size,
                              hipStream_t stream) {
    (void)in_sizes; (void)n_in; (void)out_size; (void)ws_size;
    static const int nodeN[4] = {100000, 10000, 5000, 1000};
    static const int nodeD[4] = {8, 10, 7, 3};
    // edge: srcType, dstType, numEdges   (block=0, spmt=1, crane=2, facility=3)
    static const int eST[8] = {0, 1, 0, 2, 0, 0, 1, 2};
    static const int eDT[8] = {1, 0, 2, 0, 3, 0, 3, 3};
    static const int eNE[8] = {800000, 800000, 400000, 400000, 100000, 400000, 10000, 5000};
    // proj sorted-key position for node type (block,crane,facility,spmt)
    static const int projPos[4] = {0, 3, 1, 2};
    // incoming edge-type ids per dst node type (-1 = none)
    static const int inc[4][3] = {{1, 3, 5}, {0, -1, -1}, {2, -1, -1}, {4, 6, 7}};
    static const float invk[4] = {1.f / 3.f, 1.f, 1.f, 1.f / 3.f};

    const float* xin[4]; const int* esrc[8]; const int* edst[8]; const int* batch[4];
    for (int i = 0; i < 4; ++i) xin[i] = (const float*)d_in[i];
    for (int i = 0; i < 8; ++i) { esrc[i] = (const int*)d_in[4 + 2 * i]; edst[i] = (const int*)d_in[5 + 2 * i]; }
    for (int i = 0; i < 4; ++i) batch[i] = (const int*)d_in[20 + i];
    // params pytree (sorted keys): layers(2 x 8 x [W,att_dst,att_src,bias]),
    // norms(2 x [b,g]), proj(4 sorted x [W,b]) -> leaves start at 24
    const int PBASE = 24;
    auto P = [&](int i) { return (const float*)d_in[i]; };
    auto relIdx = [&](int li, int ei) { return PBASE + li * 32 + ei * 4; };
    const int NORMB = PBASE + 64;   // 88
    const int PROJB = NORMB + 4;    // 92

    // workspace carve-up (floats)
    float* w = (float*)d_ws;
    size_t off = 0;
    auto take = [&](size_t nf) { float* p = w + off; off += (nf + 63) & ~(size_t)63; return p; };
    float* X[4]; float* ACC[4];
    for (int nt = 0; nt < 4; ++nt) X[nt] = take((size_t)nodeN[nt] * HID);
    for (int nt = 0; nt < 4; ++nt) ACC[nt] = take((size_t)nodeN[nt] * HID);
    float* HS = take((size_t)100000 * HID);
    float* HD = take((size_t)100000 * HID);
    float* AS = take((size_t)100000 * H);
    float* AD = take((size_t)100000 * H);
    float* MB = take((size_t)100000 * H);
    float* DEN = take((size_t)100000 * H);
    float* EE = take((size_t)800000 * H);
    float* CNT = take(256);
    float* OUT = (float*)d_out;

    auto fillf = [&](float* p, float v, long n) {
        int blocks = (int)((n + 255) / 256);
        if (blocks > 65535) blocks = 65535;
        fill_kernel<<<blocks, 256, 0, stream>>>(p, v, n);
    };

    // ---- input projections
    for (int nt = 0; nt < 4; ++nt) {
        const float* Wp = P(PROJB + projPos[nt] * 2);
        const float* bp = P(PROJB + projPos[nt] * 2 + 1);
        long total = (long)nodeN[nt] * HID;
        proj_kernel<<<(int)((total + 255) / 256), 256, 0, stream>>>(xin[nt], Wp, bp, X[nt],
                                                                    nodeN[nt], nodeD[nt]);
    }

    // ---- GNN layers
    for (int li = 0; li < 2; ++li) {
        for (int nt = 0; nt < 4; ++nt) fillf(ACC[nt], 0.f, (long)nodeN[nt] * HID);

        for (int ei = 0; ei < 8; ++ei) {
            int st = eST[ei], dt = eDT[ei], E = eNE[ei];
            int Ns = nodeN[st], Nd = nodeN[dt];
            const float* Wr = P(relIdx(li, ei));
            const float* attD = P(relIdx(li, ei) + 1);
            const float* attS = P(relIdx(li, ei) + 2);

            gemm128_wmma<<<(Ns + 15) / 16, 256, 0, stream>>>(X[st], Wr, HS, Ns);
            gemm128_wmma<<<(Nd + 15) / 16, 256, 0, stream>>>(X[dt], Wr, HD, Nd);
            attdot_kernel<<<(int)(((long)Ns * H + 255) / 256), 256, 0, stream>>>(HS, attS, AS, Ns);
            attdot_kernel<<<(int)(((long)Nd * H + 255) / 256), 256, 0, stream>>>(HD, attD, AD, Nd);

            fillf(MB, -__builtin_huge_valf(), (long)Nd * H);
            fillf(DEN, 0.f, (long)Nd * H);

            long eh = (long)E * H;
            edge_max_kernel<<<(int)((eh + 255) / 256), 256, 0, stream>>>(esrc[ei], edst[ei],
                                                                         AS, AD, MB, E);
            edge_exp_kernel<<<(int)((eh + 255) / 256), 256, 0, stream>>>(esrc[ei], edst[ei],
                                                                         AS, AD, MB, EE, DEN, E);
            long ec = (long)E * HID;
            edge_scatter_kernel<<<(int)((ec + 255) / 256), 256, 0, stream>>>(esrc[ei], edst[ei],
                                                                             HS, EE, DEN, ACC[dt], E);
        }

        const float* nb = P(NORMB + li * 2);      // norm beta
        const float* ng = P(NORMB + li * 2 + 1);  // norm gamma
        for (int nt = 0; nt < 4; ++nt) {
            const float* b0 = inc[nt][0] >= 0 ? P(relIdx(li, inc[nt][0]) + 3) : nullptr;
            const float* b1 = inc[nt][1] >= 0 ? P(relIdx(li, inc[nt][1]) + 3) : nullptr;
            const float* b2 = inc[nt][2] >= 0 ? P(relIdx(li, inc[nt][2]) + 3) : nullptr;
            post_kernel<<<nodeN[nt], HID, 0, stream>>>(X[nt], ACC[nt], b0, b1, b2,
                                                       invk[nt], ng, nb, nodeN[nt]);
        }
    }

    // ---- global mean pool -> [64, 512]
    fillf(OUT, 0.f, 64 * 4 * HID);
    fillf(CNT, 0.f, 256);
    for (int nt = 0; nt < 4; ++nt) {
        long total = (long)nodeN[nt] * HID;
        pool_kernel<<<(int)((total + 255) / 256), 256, 0, stream>>>(X[nt], batch[nt], OUT, CNT,
                                                                    nodeN[nt], nt);
    }
    pool_div_kernel<<<(64 * 4 * HID + 255) / 256, 256, 0, stream>>>(OUT, CNT);
}